// Resample_3642132267022
// MI455X (gfx1250) — compile-verified
//
#include <hip/hip_runtime.h>

// CDNA5 / gfx1250 wave32 WMMA polyphase resampler (4 -> 5), f32 end-to-end.
// Memory-bound (~302 MB mandatory HBM traffic @ 23.3 TB/s ~= 13 us floor;
// only ~4.7 GFLOP), so: b128 coalesced streaming through LDS, compute via
// V_WMMA_F32_16X16X4_F32 (native f32), phases padded 5->16 in the N dim.

typedef __attribute__((ext_vector_type(2))) float v2f;
typedef __attribute__((ext_vector_type(4))) float f4;
typedef __attribute__((ext_vector_type(8))) float v8f;

#define OLD_SR 4
#define NEW_SR 5
#define KW     56                 // filter taps per phase
#define T_IN   1048576
#define O_POS  262144             // conv positions emitted per row
#define OUT_T  (NEW_SR * O_POS)   // 1310720

#define PPB    128                // conv positions per pass (8 waves * 16)
#define ITERS  8                  // passes per block
#define IN_V4  144                // staged input float4s (576 floats; need 568)
#define OUT_V4 160                // staged output float4s (640 floats)
#define NKC    (KW / 4)           // 14 K-chunks of 4

__global__ __launch_bounds__(256)
void resample_wmma_f32(const float* __restrict__ x,
                       const float* __restrict__ kern,
                       float* __restrict__ out)
{
    __shared__ f4 s_in4[IN_V4];      // s_in[u] = x[gbase + u] (clamped at edges)
    __shared__ f4 s_out4[OUT_V4];    // 640 contiguous outputs for this pass
    float* const s_in  = (float*)s_in4;
    float* const s_out = (float*)s_out4;

    const int tid   = threadIdx.x;
    const int lane  = tid & 31;
    const int wave  = tid >> 5;
    const int laneM = lane & 15;     // M for A, N/phase for B & D
    const int hi    = lane >> 4;     // lower/upper half-lanes
    const int batch = blockIdx.y;
    const float* __restrict__ xrow = x + (long long)batch * T_IN;

    // ---- Filter bank as 14 WMMA B-tiles (4x16), loaded once per block.
    // B (KxN) wave32 layout: VGPR0 = row K=0 (lanes 0-15) / K=2 (lanes 16-31),
    // VGPR1 = row K=1 / K=3.  Phases 5..15 zero-padded.
    v2f bmat[NKC];
    #pragma unroll
    for (int kc = 0; kc < NKC; ++kc) {
        float bx = 0.0f, by = 0.0f;
        if (laneM < NEW_SR) {
            const int k0 = kc * 4 + 2 * hi;
            bx = kern[laneM * KW + k0];
            by = kern[laneM * KW + k0 + 1];
        }
        bmat[kc].x = bx;
        bmat[kc].y = by;
    }

    for (int it = 0; it < ITERS; ++it) {
        const int nblock = (blockIdx.x * ITERS + it) * PPB;
        // 16B-aligned global base of the staged window; element (p,k) of the
        // convolution lives at s_in[p*4 + k + 2].
        const long long gbase = (long long)nblock * OLD_SR - 28;

        if (gbase >= 0 && gbase + 4LL * IN_V4 <= T_IN) {
            // Interior pass (all but 2 passes per batch row): one b128
            // global load + one b128 LDS store per active thread.
            if (tid < IN_V4) {
                const f4* __restrict__ src = (const f4*)(xrow + gbase);
                s_in4[tid] = src[tid];
            }
        } else {
            // Edge pass: scalar loads with replicate-pad clamping.
            for (int u = tid; u < 4 * IN_V4; u += 256) {
                long long gi = gbase + u;
                gi = gi < 0 ? 0 : (gi > (long long)(T_IN - 1) ? (T_IN - 1) : gi);
                s_in[u] = xrow[gi];
            }
        }

        // Speculative prefetch of next pass's window (global_prefetch_b8).
        if (it + 1 < ITERS && tid < 10) {
            const long long pg = gbase + PPB * OLD_SR + (long long)tid * 64;
            if (pg < T_IN) __builtin_prefetch(xrow + pg, 0, 1);
        }
        __syncthreads();

        // ---- 16 positions x 16 phases per wave, K=56 as 14 chunks of 4.
        // A (16x4) wave32 layout: lanes 0-15 hold K=0,1; lanes 16-31 K=2,3.
        const int pbase = wave * 16;
        v8f c = {0.f, 0.f, 0.f, 0.f, 0.f, 0.f, 0.f, 0.f};
        #pragma unroll
        for (int kc = 0; kc < NKC; ++kc) {
            const int base = (pbase + laneM) * OLD_SR + kc * 4 + 2 * hi + 2;
            v2f a;
            a.x = s_in[base];
            a.y = s_in[base + 1];
            // (neg_a, A, neg_b, B, c_mod, C, reuse_a, reuse_b)
            c = __builtin_amdgcn_wmma_f32_16x16x4_f32(
                    false, a, false, bmat[kc], (short)0, c, false, false);
        }

        // ---- D layout: VGPR r = M=r (lanes 0-15) / M=r+8 (lanes 16-31).
        // Funnel valid phases through LDS for a contiguous b128 store region.
        if (laneM < NEW_SR) {
            #pragma unroll
            for (int r = 0; r < 8; ++r) {
                s_out[(pbase + r + 8 * hi) * NEW_SR + laneM] = c[r];
            }
        }
        __syncthreads();

        if (tid < OUT_V4) {
            f4* __restrict__ dst =
                (f4*)(out + (long long)batch * OUT_T + (long long)nblock * NEW_SR);
            dst[tid] = s_out4[tid];
        }
        __syncthreads();   // protect s_in/s_out for the next pass
    }
}

extern "C" void kernel_launch(void* const* d_in, const int* in_sizes, int n_in,
                              void* d_out, int out_size, void* d_ws, size_t ws_size,
                              hipStream_t stream) {
    (void)in_sizes; (void)n_in; (void)d_ws; (void)ws_size; (void)out_size;
    const float* x    = (const float*)d_in[0];   // [32, 1048576] f32
    const float* kern = (const float*)d_in[1];   // [5, 56] f32
    float* out        = (float*)d_out;           // [32, 1310720] f32

    dim3 grid(O_POS / (PPB * ITERS), 32);        // (256, 32)
    resample_wmma_f32<<<grid, 256, 0, stream>>>(x, kern, out);
}